// GIN_13400297963802
// MI455X (gfx1250) — compile-verified
//
#include <hip/hip_runtime.h>
#include <hip/hip_bf16.h>

typedef __attribute__((ext_vector_type(2))) float v2f;
typedef __attribute__((ext_vector_type(8))) float v8f;

#define NNODES 100000
#define NEDGES 1600000
#define IN_DIM 8
#define EMB 50

// ---------------- utility ----------------
__global__ void zero_f(float* __restrict__ p, unsigned n) {
    unsigned i = blockIdx.x * blockDim.x + threadIdx.x;
    if (i < n) p[i] = 0.0f;
}

// ---------------- precompute: per-dst degree + segment_sum(edge_feats) ----------------
__global__ void edge_pre(const int* __restrict__ dst, const float* __restrict__ efeat,
                         float* __restrict__ deg, float* __restrict__ EF, unsigned E) {
    unsigned tid = blockIdx.x * blockDim.x + threadIdx.x;
    if (tid >= E * IN_DIM) return;
    unsigned e = tid >> 3;
    unsigned k = tid & 7;
    int d = dst[e];
    if (k == 0) atomicAdd(&deg[d], 1.0f);
    atomicAdd(&EF[(unsigned)d * IN_DIM + k], efeat[e * IN_DIM + k]);
}

__global__ void make_invdeg(const float* __restrict__ deg, float* __restrict__ invdeg, unsigned n) {
    unsigned i = blockIdx.x * blockDim.x + threadIdx.x;
    if (i >= n) return;
    float d = deg[i];
    invdeg[i] = (d > 0.0f) ? (1.0f / d) : 0.0f;
}

// ---------------- node encoder: h = node_feats @ node_W + node_b ----------------
__global__ void node_enc(const float* __restrict__ nf, const float* __restrict__ Wm,
                         const float* __restrict__ b, float* __restrict__ h) {
    unsigned tid = blockIdx.x * blockDim.x + threadIdx.x;
    if (tid >= (unsigned)NNODES * EMB) return;
    unsigned n = tid / EMB;
    unsigned j = tid - n * EMB;
    float acc = b[j];
#pragma unroll
    for (int k = 0; k < IN_DIM; ++k)
        acc += nf[n * IN_DIM + k] * Wm[k * EMB + j];
    h[tid] = acc;
}

// ---------------- per-layer: agg = EF @ edge_W + deg*edge_b (== segment_sum(e)) ----------------
__global__ void init_agg(const float* __restrict__ EF, const float* __restrict__ eW,
                         const float* __restrict__ eb, const float* __restrict__ deg,
                         float* __restrict__ agg) {
    unsigned tid = blockIdx.x * blockDim.x + threadIdx.x;
    if (tid >= (unsigned)NNODES * EMB) return;
    unsigned n = tid / EMB;
    unsigned j = tid - n * EMB;
    float acc = 0.0f;
#pragma unroll
    for (int c = 0; c < IN_DIM; ++c)
        acc += EF[n * IN_DIM + c] * eW[c * EMB + j];
    agg[tid] = acc + deg[n] * eb[j];
}

// ---------------- per-layer edge pass: agg[dst] += h[src]  (64 lanes per edge, 50 active) ----------------
__global__ void edge_gather(const int* __restrict__ src, const int* __restrict__ dst,
                            const float* __restrict__ h, float* __restrict__ agg, unsigned E) {
    unsigned tid = blockIdx.x * blockDim.x + threadIdx.x;
    if (tid >= E * 64u) return;
    unsigned k = tid & 63u;
    if (k >= EMB) return;
    unsigned e = tid >> 6;
    int s = src[e];
    int d = dst[e];
    atomicAdd(&agg[(unsigned)d * EMB + k], h[(unsigned)s * EMB + k]);
}

// ---------------- combine: comb = (1+eps)*h + agg*invdeg ----------------
__global__ void combine_k(const float* __restrict__ h, const float* __restrict__ agg,
                          const float* __restrict__ invdeg, const float* __restrict__ eps,
                          int layer, float* __restrict__ comb) {
    unsigned tid = blockIdx.x * blockDim.x + threadIdx.x;
    if (tid >= (unsigned)NNODES * EMB) return;
    unsigned n = tid / EMB;
    float ev = 1.0f + eps[layer];
    comb[tid] = ev * h[tid] + agg[tid] * invdeg[n];
}

// ---------------- WMMA f32 GEMM: C[M,Ncols] = A[M,K] @ W[K,Ncols] + bias ----------------
// One wave32 per 32x16 output block (two 16x16 WMMA tiles sharing B).
// Requires M % 32 == 0 (N nodes = 100000 = 3125*32). K templated (50 or 100) so
// the unrolled main loop has unconditional, exec-mask-free loads; A operand pair
// is fetched as one aligned 8-byte v2f load.
// V_WMMA_F32_16X16X4_F32 layouts:
//   A: lanes 0-15 hold row M=lane (K=k0,k0+1); lanes 16-31 same rows (K=k0+2,k0+3)
//   B: lanes 0-15 hold col N=lane (K=k0,k0+1); lanes 16-31 same cols (K=k0+2,k0+3)
//   C/D: lane<16 -> rows 0..7 in v[0..7]; lane>=16 -> rows 8..15
template <int K>
__global__ __launch_bounds__(32) void wmma_gemm_f32(const float* __restrict__ A,
                                                    const float* __restrict__ W,
                                                    const float* __restrict__ bias,
                                                    float* __restrict__ C,
                                                    int Ncols) {
    constexpr int K4 = K & ~3;
    const int lane = threadIdx.x;
    const int half = lane >> 4;
    const int l15 = lane & 15;
    const int row0 = blockIdx.x * 32 + l15;      // first M tile
    const int col = blockIdx.y * 16 + l15;
    const int colc = (col < Ncols) ? col : (Ncols - 1);   // clamp loads, guard stores

    const float* Ap0 = A + (size_t)row0 * K + half * 2;
    const float* Ap1 = Ap0 + (size_t)16 * K;              // second M tile (row0+16)
    const float* Wp  = W + (size_t)(half * 2) * Ncols + colc;

    v8f acc0 = {};
    v8f acc1 = {};
#pragma unroll
    for (int k0 = 0; k0 < K4; k0 += 4) {
        v2f a0 = *(const v2f*)(Ap0 + k0);
        v2f a1 = *(const v2f*)(Ap1 + k0);
        v2f b;
        b.x = Wp[(size_t)k0 * Ncols];
        b.y = Wp[(size_t)(k0 + 1) * Ncols];
        acc0 = __builtin_amdgcn_wmma_f32_16x16x4_f32(false, a0, false, b,
                                                     (short)0, acc0, false, false);
        acc1 = __builtin_amdgcn_wmma_f32_16x16x4_f32(false, a1, false, b,
                                                     (short)0, acc1, false, false);
    }
    if constexpr (K4 < K) {
        const int ka = K4 + half * 2;
        v2f a0 = {0.0f, 0.0f};
        v2f a1 = {0.0f, 0.0f};
        v2f b = {0.0f, 0.0f};
        if (ka < K) {
            a0.x = Ap0[K4];
            a1.x = Ap1[K4];
            b.x = Wp[(size_t)K4 * Ncols];
        }
        if (ka + 1 < K) {
            a0.y = Ap0[K4 + 1];
            a1.y = Ap1[K4 + 1];
            b.y = Wp[(size_t)(K4 + 1) * Ncols];
        }
        acc0 = __builtin_amdgcn_wmma_f32_16x16x4_f32(false, a0, false, b,
                                                     (short)0, acc0, false, false);
        acc1 = __builtin_amdgcn_wmma_f32_16x16x4_f32(false, a1, false, b,
                                                     (short)0, acc1, false, false);
    }
    if (col < Ncols) {
        const float bv = bias[col];
        size_t cb = (size_t)(blockIdx.x * 32 + half * 8) * Ncols + col;
#pragma unroll
        for (int j = 0; j < 8; ++j) C[cb + (size_t)j * Ncols] = acc0[j] + bv;
        cb += (size_t)16 * Ncols;
#pragma unroll
        for (int j = 0; j < 8; ++j) C[cb + (size_t)j * Ncols] = acc1[j] + bv;
    }
}

// ---------------- BN stats: per-column sum and sum of squares ----------------
__global__ void bn_stats(const float* __restrict__ X, float* __restrict__ colsum,
                         float* __restrict__ colsumsq, int M, int Ncols, int rowsPerBlock) {
    int col = threadIdx.x;
    if (col >= Ncols) return;
    int r0 = blockIdx.x * rowsPerBlock;
    int r1 = r0 + rowsPerBlock;
    if (r1 > M) r1 = M;
    float s = 0.0f, ss = 0.0f;
    for (int r = r0; r < r1; ++r) {
        float v = X[(long long)r * Ncols + col];
        s += v;
        ss += v * v;
    }
    atomicAdd(&colsum[col], s);
    atomicAdd(&colsumsq[col], ss);
}

__global__ void bn_finalize(const float* __restrict__ colsum, const float* __restrict__ colsumsq,
                            const float* __restrict__ g, const float* __restrict__ be,
                            float* __restrict__ scale, float* __restrict__ shift,
                            int Ncols, float invM) {
    int c = threadIdx.x;
    if (c >= Ncols) return;
    float mu = colsum[c] * invM;
    float var = colsumsq[c] * invM - mu * mu;
    float sc = g[c] * rsqrtf(var + 1e-5f);
    scale[c] = sc;
    shift[c] = be[c] - mu * sc;
}

__global__ void bn_apply_relu(const float* __restrict__ X, const float* __restrict__ scale,
                              const float* __restrict__ shift, float* __restrict__ Y,
                              unsigned total, int Ncols) {
    unsigned i = blockIdx.x * blockDim.x + threadIdx.x;
    if (i >= total) return;
    unsigned col = i % (unsigned)Ncols;
    float v = X[i] * scale[col] + shift[col];
    Y[i] = v > 0.0f ? v : 0.0f;
}

extern "C" void kernel_launch(void* const* d_in, const int* in_sizes, int n_in,
                              void* d_out, int out_size, void* d_ws, size_t ws_size,
                              hipStream_t stream) {
    const int*   src        = (const int*)d_in[0];
    const int*   dst        = (const int*)d_in[1];
    const float* node_feats = (const float*)d_in[2];
    const float* edge_feats = (const float*)d_in[3];
    const float* node_W     = (const float*)d_in[4];
    const float* node_b     = (const float*)d_in[5];
    const float* edge_W     = (const float*)d_in[6];
    const float* edge_b     = (const float*)d_in[7];
    const float* eps        = (const float*)d_in[8];
    const float* W1         = (const float*)d_in[9];
    const float* b1         = (const float*)d_in[10];
    const float* g1         = (const float*)d_in[11];
    const float* be1        = (const float*)d_in[12];
    const float* W2         = (const float*)d_in[13];
    const float* b2         = (const float*)d_in[14];
    const float* g2         = (const float*)d_in[15];
    const float* be2        = (const float*)d_in[16];
    const float* pred_W     = (const float*)d_in[17];
    const float* pred_b     = (const float*)d_in[18];
    float* out = (float*)d_out;

    const unsigned N = NNODES, E = NEDGES;
    float* ws       = (float*)d_ws;
    float* h        = ws;                         // N*50
    float* agg      = h + (size_t)N * EMB;        // N*50
    float* comb     = agg + (size_t)N * EMB;      // N*50 (also x2)
    float* x1       = comb + (size_t)N * EMB;     // N*100
    float* EF       = x1 + (size_t)N * 2 * EMB;   // N*8
    float* deg      = EF + (size_t)N * IN_DIM;    // N
    float* invdeg   = deg + N;                    // N
    float* colsum   = invdeg + N;                 // 128
    float* colsumsq = colsum + 128;               // 128
    float* scale    = colsumsq + 128;             // 128
    float* shift    = scale + 128;                // 128

    const int TB = 256;
    // zero EF + deg (contiguous N*9 floats)
    zero_f<<<(N * 9 + TB - 1) / TB, TB, 0, stream>>>(EF, N * 9);
    // precompute degree + segment_sum(edge_feats, dst)
    edge_pre<<<(E * IN_DIM + TB - 1) / TB, TB, 0, stream>>>(dst, edge_feats, deg, EF, E);
    // node encoder
    node_enc<<<(N * EMB + TB - 1) / TB, TB, 0, stream>>>(node_feats, node_W, node_b, h);
    make_invdeg<<<(N + TB - 1) / TB, TB, 0, stream>>>(deg, invdeg, N);

    const int rowsPerBlock = 256;
    const int statBlocks = (N + rowsPerBlock - 1) / rowsPerBlock;

    for (int layer = 0; layer < 2; ++layer) {
        // agg = segment_sum(e) = EF @ edge_W + deg*edge_b
        init_agg<<<(N * EMB + TB - 1) / TB, TB, 0, stream>>>(EF, edge_W, edge_b, deg, agg);
        // agg += segment_sum(h[src])
        edge_gather<<<(E * 64u + TB - 1) / TB, TB, 0, stream>>>(src, dst, h, agg, E);
        // comb = (1+eps)*h + agg*invdeg
        combine_k<<<(N * EMB + TB - 1) / TB, TB, 0, stream>>>(h, agg, invdeg, eps, layer, comb);

        // x1 = comb @ W1[layer] + b1[layer]   (50 -> 100)
        {
            dim3 grid(N / 32, (2 * EMB + 15) / 16);
            wmma_gemm_f32<EMB><<<grid, 32, 0, stream>>>(comb, W1 + (size_t)layer * EMB * 2 * EMB,
                                                        b1 + (size_t)layer * 2 * EMB, x1, 2 * EMB);
        }
        zero_f<<<1, 256, 0, stream>>>(colsum, 256);
        bn_stats<<<statBlocks, 128, 0, stream>>>(x1, colsum, colsumsq, N, 2 * EMB, rowsPerBlock);
        bn_finalize<<<1, 128, 0, stream>>>(colsum, colsumsq, g1 + (size_t)layer * 2 * EMB,
                                           be1 + (size_t)layer * 2 * EMB, scale, shift,
                                           2 * EMB, 1.0f / (float)N);
        bn_apply_relu<<<(N * 2 * EMB + TB - 1) / TB, TB, 0, stream>>>(x1, scale, shift, x1,
                                                                      N * 2 * EMB, 2 * EMB);

        // x2 = x1 @ W2[layer] + b2[layer]   (100 -> 50), into comb buffer
        {
            dim3 grid(N / 32, (EMB + 15) / 16);
            wmma_gemm_f32<2 * EMB><<<grid, 32, 0, stream>>>(x1, W2 + (size_t)layer * 2 * EMB * EMB,
                                                            b2 + (size_t)layer * EMB, comb, EMB);
        }
        zero_f<<<1, 256, 0, stream>>>(colsum, 256);
        bn_stats<<<statBlocks, 128, 0, stream>>>(comb, colsum, colsumsq, N, EMB, rowsPerBlock);
        bn_finalize<<<1, 128, 0, stream>>>(colsum, colsumsq, g2 + (size_t)layer * EMB,
                                           be2 + (size_t)layer * EMB, scale, shift,
                                           EMB, 1.0f / (float)N);
        bn_apply_relu<<<(N * EMB + TB - 1) / TB, TB, 0, stream>>>(comb, scale, shift, h,
                                                                  N * EMB, EMB);
    }

    // out = h @ pred_W + pred_b   (50 -> 112)
    {
        dim3 grid(N / 32, (112 + 15) / 16);
        wmma_gemm_f32<EMB><<<grid, 32, 0, stream>>>(h, pred_W, pred_b, out, 112);
    }
}